// LowLevelGRU_14207751815708
// MI455X (gfx1250) — compile-verified
//
#include <hip/hip_runtime.h>
#include <hip/hip_bf16.h>

#define T_STEPS 4096
#define X_DIM   256
#define H_DIM   2048
#define Y_DIM   256
#define NBLK_SCAN 128

typedef __attribute__((ext_vector_type(2))) float v2f;
typedef __attribute__((ext_vector_type(8))) float v8f;

// ---------------------------------------------------------------------------
// Device-scope barrier for a resident grid. `target` is a monotone barrier
// index so there is no read-generation-before-arrive race.
// ---------------------------------------------------------------------------
__device__ __forceinline__ void grid_barrier(int* count, volatile int* gen,
                                             int nblk, int target) {
    __threadfence();            // release: my stores visible at device scope
    __syncthreads();
    if (threadIdx.x == 0) {
        if (atomicAdd(count, 1) == nblk - 1) {
            atomicExch(count, 0);
            __threadfence();
            atomicAdd((int*)gen, 1);       // gen becomes `target`
        } else {
            while (*gen < target) __builtin_amdgcn_s_sleep(2);
        }
    }
    __syncthreads();
    __threadfence();            // acquire: invalidate stale L0 lines
}

// ---------------------------------------------------------------------------
// init: h_cur = h0, zero barrier state
// ---------------------------------------------------------------------------
__global__ void init_k(const float* __restrict__ h0, float* __restrict__ h_cur,
                       int* __restrict__ bar) {
    int i = blockIdx.x * 256 + threadIdx.x;
    if (i < H_DIM) h_cur[i] = h0[i];
    if (i < 2) bar[i] = 0;
}

// ---------------------------------------------------------------------------
// pre_gemm: G[gate] = x @ W[gate][:256,:] + b[gate]   (4096x2048, f32 WMMA)
// One wave per 16x16 tile, K-loop of 64 x V_WMMA_F32_16X16X4_F32.
// ---------------------------------------------------------------------------
__global__ __launch_bounds__(128) void pre_gemm_k(
    const float* __restrict__ x,
    const float* __restrict__ Wu, const float* __restrict__ Wr,
    const float* __restrict__ Wc,
    const float* __restrict__ bu, const float* __restrict__ br,
    const float* __restrict__ bc,
    float* __restrict__ Gu, float* __restrict__ Gr, float* __restrict__ Gc) {
    const int gate = blockIdx.y;
    const float* W = (gate == 0) ? Wu : (gate == 1) ? Wr : Wc;
    const float* b = (gate == 0) ? bu : (gate == 1) ? br : bc;
    float*       G = (gate == 0) ? Gu : (gate == 1) ? Gr : Gc;

    const int wave = threadIdx.x >> 5;
    const int lane = threadIdx.x & 31;
    const int tile = blockIdx.x * 4 + wave;        // 0..32767
    const int mt = tile >> 7;                      // 256 row tiles
    const int nt = tile & 127;                     // 128 col tiles

    const int mA  = mt * 16 + (lane & 15);         // A: lane -> row M
    const int kHi = (lane >> 4) * 2;               // hi half-wave -> K=2,3
    const int n   = nt * 16 + (lane & 15);         // B/C/D: lane -> col N

    v8f acc = {0.f, 0.f, 0.f, 0.f, 0.f, 0.f, 0.f, 0.f};
    const float* xrow = x + (size_t)mA * X_DIM;
    for (int k0 = 0; k0 < X_DIM; k0 += 4) {
        v2f a, bf;
        a.x  = xrow[k0 + kHi];
        a.y  = xrow[k0 + kHi + 1];
        bf.x = W[(size_t)(k0 + kHi) * H_DIM + n];
        bf.y = W[(size_t)(k0 + kHi + 1) * H_DIM + n];
        acc = __builtin_amdgcn_wmma_f32_16x16x4_f32(
            false, a, false, bf, (short)0, acc, false, false);
    }
    const float bias = b[n];
    const int mrow = mt * 16 + (lane >> 4) * 8;    // C/D: VGPR r -> row r/r+8
    #pragma unroll
    for (int r = 0; r < 8; ++r)
        G[(size_t)(mrow + r) * H_DIM + n] = acc[r] + bias;
}

// ---------------------------------------------------------------------------
// scan: persistent kernel, NBLK_SCAN resident blocks, 2 device barriers/step.
// pass1: u = sigm(Gu + h@Wu_h), hr = h * sigm(Gr + h@Wr_h)
// pass2: c = tanh(Gc + hr@Wc_h); h = c*u + h*(1-u); store h into Hbuf.
// Weight streaming is L2-resident; b128 loads: 1 vmem op feeds 4 FMAs.
// ---------------------------------------------------------------------------
__global__ __launch_bounds__(256) void scan_k(
    const float* __restrict__ Gu, const float* __restrict__ Gr,
    const float* __restrict__ Gc,
    const float* __restrict__ Wu, const float* __restrict__ Wr,
    const float* __restrict__ Wc,
    float* __restrict__ h_cur, float* __restrict__ u_vec,
    float* __restrict__ hr_vec, float* __restrict__ Hbuf,
    int* __restrict__ barCount, int* __restrict__ barGen) {
    __shared__ float  lds_v[H_DIM];        // 8 KB: h (pass1) / h*r (pass2)
    __shared__ float4 lds_red4[256];       // 4 KB: partial dot sums
    const float* lds_red = (const float*)lds_red4;
    const int tid = threadIdx.x;
    const int bid = blockIdx.x;
    const float* Wuh = Wu + (size_t)X_DIM * H_DIM;
    const float* Wrh = Wr + (size_t)X_DIM * H_DIM;
    const float* Wch = Wc + (size_t)X_DIM * H_DIM;
    int bar = 0;

    #pragma clang loop unroll(disable)
    for (int t = 0; t < T_STEPS; ++t) {
        // ---- pass 1: gates u (blocks 0..63) and r (blocks 64..127) ----
        for (int i = tid; i < H_DIM; i += 256) lds_v[i] = h_cur[i];
        __syncthreads();
        {
            const int half = bid >> 6;                 // 0:u  1:r
            const int jb   = (bid & 63) * 32;          // 32 outputs per block
            const int grp  = tid & 7;                  // 8 groups of 4 columns
            const int seg  = tid >> 3;                 // 32 segments of 64 rows
            const int j4   = jb + grp * 4;
            const float* Wh = half ? Wrh : Wuh;
            const int i0 = seg * 64;
            float4 acc = {0.f, 0.f, 0.f, 0.f};
            #pragma unroll 4
            for (int i = i0; i < i0 + 64; ++i) {
                const float hv = lds_v[i];
                const float4 w =
                    *(const float4*)&Wh[(size_t)i * H_DIM + j4];
                acc.x += hv * w.x; acc.y += hv * w.y;
                acc.z += hv * w.z; acc.w += hv * w.w;
            }
            lds_red4[tid] = acc;
            __syncthreads();
            if (tid < 32) {                            // j = jb + tid
                const int g = tid >> 2, c = tid & 3;
                float s = 0.f;
                #pragma unroll
                for (int k = 0; k < 32; ++k)
                    s += lds_red[(g + 8 * k) * 4 + c];
                const int j = jb + tid;
                const float* Gh = half ? Gr : Gu;
                const float z = Gh[(size_t)t * H_DIM + j] + s;
                const float sig = 1.f / (1.f + __expf(-z));
                if (half) hr_vec[j] = lds_v[j] * sig;
                else      u_vec[j]  = sig;
                if (t + 1 < T_STEPS)                   // hide next-step G fetch
                    __builtin_prefetch(&Gh[(size_t)(t + 1) * H_DIM + j], 0, 3);
            }
        }
        grid_barrier(barCount, barGen, NBLK_SCAN, ++bar);

        // ---- pass 2: candidate + state update (all 128 blocks) ----
        for (int i = tid; i < H_DIM; i += 256) lds_v[i] = hr_vec[i];
        __syncthreads();
        {
            const int jb  = bid * 16;                  // 16 outputs per block
            const int grp = tid & 3;                   // 4 groups of 4 columns
            const int seg = tid >> 2;                  // 64 segments of 32 rows
            const int j4  = jb + grp * 4;
            const int i0 = seg * 32;
            float4 acc = {0.f, 0.f, 0.f, 0.f};
            #pragma unroll 4
            for (int i = i0; i < i0 + 32; ++i) {
                const float hv = lds_v[i];
                const float4 w =
                    *(const float4*)&Wch[(size_t)i * H_DIM + j4];
                acc.x += hv * w.x; acc.y += hv * w.y;
                acc.z += hv * w.z; acc.w += hv * w.w;
            }
            lds_red4[tid] = acc;
            __syncthreads();
            if (tid < 16) {                            // j = jb + tid
                const int g = tid >> 2, c = tid & 3;
                float s = 0.f;
                #pragma unroll
                for (int k = 0; k < 64; ++k)
                    s += lds_red[(g + 4 * k) * 4 + c];
                const int j = jb + tid;
                const float cc = tanhf(Gc[(size_t)t * H_DIM + j] + s);
                const float u = u_vec[j];
                const float hold = h_cur[j];
                const float hnew = cc * u + hold * (1.f - u);
                h_cur[j] = hnew;
                Hbuf[(size_t)t * H_DIM + j] = hnew;
                if (t + 1 < T_STEPS)
                    __builtin_prefetch(&Gc[(size_t)(t + 1) * H_DIM + j], 0, 3);
            }
        }
        grid_barrier(barCount, barGen, NBLK_SCAN, ++bar);
    }
}

// ---------------------------------------------------------------------------
// post_gemm: ys = Hbuf @ Wh_y + by   (4096x256, K=2048, f32 WMMA)
// ---------------------------------------------------------------------------
__global__ __launch_bounds__(128) void post_gemm_k(
    const float* __restrict__ Hbuf, const float* __restrict__ Why,
    const float* __restrict__ by, float* __restrict__ ys) {
    const int wave = threadIdx.x >> 5;
    const int lane = threadIdx.x & 31;
    const int tile = blockIdx.x * 4 + wave;        // 0..4095
    const int mt = tile >> 4;                      // 256 row tiles
    const int nt = tile & 15;                      // 16 col tiles

    const int mA  = mt * 16 + (lane & 15);
    const int kHi = (lane >> 4) * 2;
    const int n   = nt * 16 + (lane & 15);

    v8f acc = {0.f, 0.f, 0.f, 0.f, 0.f, 0.f, 0.f, 0.f};
    const float* hrow = Hbuf + (size_t)mA * H_DIM;
    for (int k0 = 0; k0 < H_DIM; k0 += 4) {
        v2f a, bf;
        a.x  = hrow[k0 + kHi];
        a.y  = hrow[k0 + kHi + 1];
        bf.x = Why[(size_t)(k0 + kHi) * Y_DIM + n];
        bf.y = Why[(size_t)(k0 + kHi + 1) * Y_DIM + n];
        acc = __builtin_amdgcn_wmma_f32_16x16x4_f32(
            false, a, false, bf, (short)0, acc, false, false);
    }
    const float bias = by[n];
    const int mrow = mt * 16 + (lane >> 4) * 8;
    #pragma unroll
    for (int r = 0; r < 8; ++r)
        ys[(size_t)(mrow + r) * Y_DIM + n] = acc[r] + bias;
}

// ---------------------------------------------------------------------------
// h_fin copy (second tuple output)
// ---------------------------------------------------------------------------
__global__ void hfin_k(const float* __restrict__ h_cur, float* __restrict__ out) {
    int i = blockIdx.x * 256 + threadIdx.x;
    if (i < H_DIM) out[i] = h_cur[i];
}

// ---------------------------------------------------------------------------
// Host launcher
// ---------------------------------------------------------------------------
extern "C" void kernel_launch(void* const* d_in, const int* in_sizes, int n_in,
                              void* d_out, int out_size, void* d_ws,
                              size_t ws_size, hipStream_t stream) {
    (void)in_sizes; (void)n_in; (void)out_size; (void)ws_size;
    const float* x   = (const float*)d_in[0];  // [1,4096,256]
    const float* h0  = (const float*)d_in[1];  // [2048]
    const float* Wc  = (const float*)d_in[2];  // [2304,2048]
    const float* Wu  = (const float*)d_in[3];
    const float* Wr  = (const float*)d_in[4];
    const float* bc  = (const float*)d_in[5];
    const float* bu  = (const float*)d_in[6];
    const float* br  = (const float*)d_in[7];
    const float* Why = (const float*)d_in[8];  // [2048,256]
    const float* by  = (const float*)d_in[9];

    float* ys   = (float*)d_out;                       // [4096,256]
    float* hfin = ys + (size_t)T_STEPS * Y_DIM;        // [2048]

    // workspace layout (~134.25 MB)
    float* Gu     = (float*)d_ws;                      // [4096,2048]
    float* Gr     = Gu + (size_t)T_STEPS * H_DIM;
    float* Gc     = Gr + (size_t)T_STEPS * H_DIM;
    float* Hbuf   = Gc + (size_t)T_STEPS * H_DIM;      // [4096,2048]
    float* h_cur  = Hbuf + (size_t)T_STEPS * H_DIM;    // [2048]
    float* u_vec  = h_cur + H_DIM;
    float* hr_vec = u_vec + H_DIM;
    int*   bar    = (int*)(hr_vec + H_DIM);            // count, gen

    init_k<<<8, 256, 0, stream>>>(h0, h_cur, bar);
    pre_gemm_k<<<dim3(8192, 3), 128, 0, stream>>>(x, Wu, Wr, Wc, bu, br, bc,
                                                  Gu, Gr, Gc);
    scan_k<<<NBLK_SCAN, 256, 0, stream>>>(Gu, Gr, Gc, Wu, Wr, Wc, h_cur, u_vec,
                                          hr_vec, Hbuf, bar, bar + 1);
    post_gemm_k<<<1024, 128, 0, stream>>>(Hbuf, Why, by, ys);
    hfin_k<<<8, 256, 0, stream>>>(h_cur, hfin);
}